// CCModel_37074157699306
// MI455X (gfx1250) — compile-verified
//
#include <hip/hip_runtime.h>

typedef __attribute__((ext_vector_type(2))) float v2f;
typedef __attribute__((ext_vector_type(8))) float v8f;

namespace {
constexpr int kNT     = 8192;   // samples per group
constexpr int kOutLen = 515;    // 2*nlag + 3
constexpr int kTiles  = 31;     // 31 tiles x 17 lags cover j in [0,515)
constexpr int kIters  = 257;    // TB = -32 + 32*i sweeps all t with K-stride-8 Hankels
constexpr int kChunk  = 64;     // sweep iterations per LDS refill
constexpr int kD2Win  = 32 * kChunk + 40;   // 2088 floats of d2 staged per chunk
constexpr int kD1Back = 256;                // d1 window starts at tbBase - 256
constexpr int kD1Win  = 2576;               // covers [tbBase-256, tbBase+2320)
}

// out[g,j] = sum_t d1[g, t + j - 257] * d2[g, t]
// WMMA cell (m,n) of tile tau accumulates lag (-249 + 17*tau) + m - n over t = n (mod 8).
__launch_bounds__(256, 1)
__global__ void xcorr_wmma_kernel(const float* __restrict__ d1,
                                  const float* __restrict__ d2,
                                  float* __restrict__ out) {
  __shared__ float sD2[kD2Win];
  __shared__ float sD1[kD1Win];
  __shared__ float sTile[32 * 256];   // one 16x16 f32 accumulator tile per (wave, tt)

  const int g    = blockIdx.x;
  const int tid  = threadIdx.x;
  const int lane = tid & 31;
  const int wave = tid >> 5;

  const float* __restrict__ g1 = d1 + (size_t)g * kNT;
  const float* __restrict__ g2 = d2 + (size_t)g * kNT;

  // WMMA f32 operand lane mapping (wave32):
  //   A 16x4 : lane&15 = M row, half-wave holds K = {0,1} or {2,3}
  //   B 4x16 : lane&15 = N col, half-wave holds K = {0,1} or {2,3}
  const int nCol = lane & 15;
  const int kb   = (lane >> 4) * 2;

  const int bBase = nCol + 8 * kb;           // + 32*li into sD2 (.y at +8)
  int aBase[4];
#pragma unroll
  for (int tt = 0; tt < 4; ++tt) {
    int tau = wave * 4 + tt;
    if (tau >= kTiles) tau = wave * 4;       // wave 7's 4th tile: harmless duplicate, never stored
    const int dT = -249 + 17 * tau;          // cell lag = dT + m - n
    aBase[tt] = kD1Back + dT + nCol + 8 * kb;
  }

  v8f acc[4] = {};

  for (int cs = 0; cs < kIters; cs += kChunk) {
    const int iters  = (kIters - cs) < kChunk ? (kIters - cs) : kChunk;
    const int tbBase = -32 + 32 * cs;

    __syncthreads();   // previous chunk fully consumed before restage
    for (int o = tid; o < kD2Win; o += 256) {
      const int gi = tbBase + o;
      sD2[o] = (gi >= 0 && gi < kNT) ? g2[gi] : 0.0f;
    }
    for (int o = tid; o < kD1Win; o += 256) {
      const int gi = tbBase - kD1Back + o;
      sD1[o] = (gi >= 0 && gi < kNT) ? g1[gi] : 0.0f;
    }
    __syncthreads();

    for (int li = 0; li < iters; ++li) {
      const int sh = 32 * li;
      v2f b;
      b.x = sD2[bBase + sh];
      b.y = sD2[bBase + sh + 8];
#pragma unroll
      for (int tt = 0; tt < 4; ++tt) {
        v2f a;
        a.x = sD1[aBase[tt] + sh];
        a.y = sD1[aBase[tt] + sh + 8];
        // 8 args: (neg_a, A, neg_b, B, c_mod, C, reuse_a, reuse_b)
        acc[tt] = __builtin_amdgcn_wmma_f32_16x16x4_f32(
            false, a, false, b, (short)0, acc[tt], false, false);
      }
    }
  }

  // ---- write-out: dump tiles, reduce 8 diagonal cells per lag ----
  const int mBase = (lane >> 4) * 8;   // C/D layout: VGPR r -> M = r + 8*(lane>=16), N = lane&15
#pragma unroll
  for (int tt = 0; tt < 4; ++tt) {
    const int tau = wave * 4 + tt;
    if (tau < kTiles) {
      float* scr = sTile + (wave * 4 + tt) * 256;
#pragma unroll
      for (int r = 0; r < 8; ++r)
        scr[(mBase + r) * 16 + nCol] = acc[tt][r];
    }
  }
  __syncthreads();

#pragma unroll
  for (int tt = 0; tt < 4; ++tt) {
    const int tau = wave * 4 + tt;
    if (tau < kTiles && lane < 17) {
      const int j = 17 * tau + lane;           // j = 17*tau + 8 + delta, delta = lane - 8
      if (j < kOutLen) {
        const int delta = lane - 8;
        const int n0 = delta < 0 ? -delta : 0; // 8 cells cover all residues t mod 8
        const float* scr = sTile + (wave * 4 + tt) * 256;
        float s = 0.0f;
#pragma unroll
        for (int q = 0; q < 8; ++q) {
          const int nn = n0 + q;
          s += scr[(nn + delta) * 16 + nn];
        }
        out[(size_t)g * kOutLen + j] = s;
      }
    }
  }
}

extern "C" void kernel_launch(void* const* d_in, const int* in_sizes, int n_in,
                              void* d_out, int out_size, void* d_ws, size_t ws_size,
                              hipStream_t stream) {
  const float* data1 = (const float*)d_in[0];
  const float* data2 = (const float*)d_in[1];
  float* out = (float*)d_out;
  const int groups = in_sizes[0] / kNT;   // nb*nc = 2048
  xcorr_wmma_kernel<<<dim3(groups), dim3(256), 0, stream>>>(data1, data2, out);
}